// UserEncoder_25632364822905
// MI455X (gfx1250) — compile-verified
//
#include <hip/hip_runtime.h>
#include <math.h>

// Sizes fixed by the reference.
#define B_SZ 512
#define L_SZ 200   // history length after add_pad
#define D_SZ 400   // news embedding dim
#define H_SZ 200   // attention hidden dim
#define G_SZ 199   // output rows per batch (L + 1 - MIN_LOG)

typedef __attribute__((ext_vector_type(2))) float v2f;
typedef __attribute__((ext_vector_type(4))) float v4f;
typedef __attribute__((ext_vector_type(8))) float v8f;

// Branchless tanh: sign(x) * (1 - e^{-2|x|}) / (1 + e^{-2|x|}).
// Straight-line v_exp_f32 + v_rcp_f32 path, no EXEC divergence (unlike libm
// tanhf, whose two-sided branchy expansion serializes both paths per wave).
__device__ __forceinline__ float tanh_fast(float x) {
  const float t = __expf(-2.0f * fabsf(x));
  const float y = (1.0f - t) * __frcp_rn(1.0f + t);
  return copysignf(y, x);
}

// ---------------------------------------------------------------------------
// Phase 1: s[b,l] = tanh(vec[b,l,:] @ W1^T + b1) @ W2^T + b2
// GEMM (B*L=102400, K=400) x (K=400, H=200) via V_WMMA_F32_16X16X4_F32.
// One wave per 16-row M tile; 13 N tiles (H padded to 208 via clamped loads);
// 100 K steps of K=4 per tile.
//
// f32 WMMA fragment layout (wave32):
//   A (16x4):  lane holds M = lane%16;  VGPR v holds K = v + 2*(lane/16)
//   B (4x16):  lane holds N = lane%16;  VGPR v holds K = v + 2*(lane/16)
//   C (16x16): lane holds N = lane%16;  VGPR j holds M = j + 8*(lane/16)
// ---------------------------------------------------------------------------
__global__ __launch_bounds__(256)
void ue_scores_wmma(const float* __restrict__ log_vec,
                    const float* __restrict__ pad,
                    const float* __restrict__ W1,
                    const float* __restrict__ b1,
                    const float* __restrict__ W2,
                    const float* __restrict__ b2,
                    float* __restrict__ s_out) {
  const int lane = threadIdx.x & 31;
  const int wave = threadIdx.x >> 5;
  const int half = lane >> 4;   // selects K pair {2*half, 2*half+1}
  const int mrow = lane & 15;

  const int m0 = (blockIdx.x * 8 + wave) * 16;   // 800 blocks * 8 waves * 16 = 102400

  // Per-lane A row pointer: row r of vec = pad (l==0) or log_vec[b, l-1, :]
  const int r = m0 + mrow;
  const int b = r / L_SZ;
  const int l = r - b * L_SZ;
  const float* arow = (l == 0) ? pad
                               : (log_vec + ((size_t)b * L_SZ + (l - 1)) * D_SZ);

  float partial[8];
#pragma unroll
  for (int j = 0; j < 8; ++j) partial[j] = 0.0f;

  for (int nt = 0; nt < 13; ++nt) {
    const int h = nt * 16 + mrow;          // hidden-unit column handled by this lane
    const bool hvalid = (h < H_SZ);
    const int hc = hvalid ? h : 0;         // clamp so loads are uniform-control-flow
    const float* brow = W1 + (size_t)hc * D_SZ;

    v8f acc = {0.f, 0.f, 0.f, 0.f, 0.f, 0.f, 0.f, 0.f};
#pragma unroll 4
    for (int kt = 0; kt < 100; ++kt) {
      const int k = kt * 4 + half * 2;     // consecutive K pair -> 8-byte load
      const v2f a  = *(const v2f*)(arow + k);
      const v2f bb = *(const v2f*)(brow + k);
      acc = __builtin_amdgcn_wmma_f32_16x16x4_f32(
          /*neg_a=*/false, a, /*neg_b=*/false, bb,
          /*c_mod=*/(short)0, acc, /*reuse_a=*/false, /*reuse_b=*/false);
    }
    if (hvalid) {
      const float w2h = W2[h];
      const float bh  = b1[h];
#pragma unroll
      for (int j = 0; j < 8; ++j)
        partial[j] += tanh_fast(acc[j] + bh) * w2h;
    }
  }

  // Reduce the H contributions across the 16 lanes of each half-wave.
#pragma unroll
  for (int off = 1; off < 16; off <<= 1) {
#pragma unroll
    for (int j = 0; j < 8; ++j)
      partial[j] += __shfl_xor(partial[j], off, 16);
  }

  if (mrow == 0) {                          // lanes 0 and 16
    const float bias2 = b2[0];
#pragma unroll
    for (int j = 0; j < 8; ++j)
      s_out[m0 + half * 8 + j] = partial[j] + bias2;
  }
}

// ---------------------------------------------------------------------------
// Phase 2: online prefix softmax + weighted accumulation (flash-style).
// out[b,g,:] = (sum_{l<=g+1} e^{s_l} vec[b,l,:]) / (sum_{l<=g+1} e^{s_l})
// One block per b; 100 active lanes x v4f cover D=400. Streaming pass:
// read vec once, NT-store out once -> bandwidth bound (~330 MB @ 23.3 TB/s).
// Zero mask rows (mask applies along g) -> uniform mean over all 200 docs.
// ---------------------------------------------------------------------------
__global__ __launch_bounds__(128)
void ue_prefix_softmax(const float* __restrict__ log_vec,
                       const float* __restrict__ log_mask,
                       const float* __restrict__ pad,
                       const float* __restrict__ s_buf,
                       float* __restrict__ out) {
  const int b = blockIdx.x;
  const int tid = threadIdx.x;

  __shared__ float sS[L_SZ];
  __shared__ int anyZero;
  if (tid == 0) anyZero = 0;
  for (int i = tid; i < L_SZ; i += 128) sS[i] = s_buf[b * L_SZ + i];
  __syncthreads();

  int zloc = 0;
  for (int i = tid; i < G_SZ; i += 128)
    if (log_mask[(size_t)b * L_SZ + i] == 0.0f) zloc = 1;
  if (zloc) atomicOr(&anyZero, 1);
  __syncthreads();

  const bool active = tid < 100;
  const int d = tid * 4;
  const float* vecb = log_vec + (size_t)b * L_SZ * D_SZ;
  float* outb = out + (size_t)b * G_SZ * D_SZ;

  v4f acc = {0.f, 0.f, 0.f, 0.f};          // running numerator
  float m = -INFINITY, Z = 0.f;            // running max / denominator

  for (int li = 0; li < L_SZ; ++li) {
    const float s = sS[li];
    const float mn = fmaxf(m, s);
    const float scale = __expf(m - mn);    // 0 on first iter (m = -inf)
    const float e = __expf(s - mn);
    Z = Z * scale + e;
    m = mn;
    if (active) {
      const float* row = (li == 0) ? (pad + d)
                                   : (vecb + (size_t)(li - 1) * D_SZ + d);
      const v4f v = *(const v4f*)row;
      acc.x = fmaf(acc.x, scale, e * v.x);
      acc.y = fmaf(acc.y, scale, e * v.y);
      acc.z = fmaf(acc.z, scale, e * v.z);
      acc.w = fmaf(acc.w, scale, e * v.w);
      if (li >= 1) {                        // emit g = li - 1
        const float inv = 1.0f / Z;
        v4f o;
        o.x = acc.x * inv; o.y = acc.y * inv; o.z = acc.z * inv; o.w = acc.w * inv;
        __builtin_nontemporal_store(o, (v4f*)(outb + (size_t)(li - 1) * D_SZ + d));
      }
    }
  }

  // Rare fixup: zero mask row g -> softmax over an all -1e8 row = uniform mean.
  if (anyZero && active) {
    v4f msum = {0.f, 0.f, 0.f, 0.f};
    for (int li = 0; li < L_SZ; ++li) {
      const float* row = (li == 0) ? (pad + d)
                                   : (vecb + (size_t)(li - 1) * D_SZ + d);
      const v4f v = *(const v4f*)row;
      msum.x += v.x; msum.y += v.y; msum.z += v.z; msum.w += v.w;
    }
    const float invL = 1.0f / (float)L_SZ;
    msum.x *= invL; msum.y *= invL; msum.z *= invL; msum.w *= invL;
    for (int g = 0; g < G_SZ; ++g) {
      if (log_mask[(size_t)b * L_SZ + g] == 0.0f) {
        float* o = outb + (size_t)g * D_SZ + d;
        o[0] = msum.x; o[1] = msum.y; o[2] = msum.z; o[3] = msum.w;
      }
    }
  }
}

// ---------------------------------------------------------------------------
extern "C" void kernel_launch(void* const* d_in, const int* in_sizes, int n_in,
                              void* d_out, int out_size, void* d_ws, size_t ws_size,
                              hipStream_t stream) {
  const float* log_vec  = (const float*)d_in[0];  // (512, 200, 400)
  const float* log_mask = (const float*)d_in[1];  // (512, 200)
  const float* pad      = (const float*)d_in[2];  // (1, 400)
  const float* W1       = (const float*)d_in[3];  // (200, 400)
  const float* b1       = (const float*)d_in[4];  // (200,)
  const float* W2       = (const float*)d_in[5];  // (1, 200)
  const float* b2       = (const float*)d_in[6];  // (1,)
  float* out  = (float*)d_out;                    // (512, 199, 400)
  float* s_buf = (float*)d_ws;                    // 512*200 floats = 400 KB scratch

  (void)in_sizes; (void)n_in; (void)out_size; (void)ws_size;

  // Phase 1: 800 blocks * 8 waves * 16 rows = 102400 = B*L score rows.
  ue_scores_wmma<<<800, 256, 0, stream>>>(log_vec, pad, W1, b1, W2, b2, s_buf);
  // Phase 2: one block per batch element.
  ue_prefix_softmax<<<B_SZ, 128, 0, stream>>>(log_vec, log_mask, pad, s_buf, out);
}